// GraphAttnBias_35132832481421
// MI455X (gfx1250) — compile-verified
//
#include <hip/hip_runtime.h>
#include <hip/hip_bf16.h>

// ---------------------------------------------------------------------------
// Graphormer attention-bias, fused for MI455X (gfx1250, wave32, WMMA).
//
// eb[b,n,m,h] = (1/(3*sp)) * sum_{dk} ei_sum[b,n,m,dk] * W[dk,h]   (K = 160)
// computed transposed per tile: C[h, m] so output stores are j-contiguous.
// Matrix op: V_WMMA_F32_16X16X4_F32 (fp32-exact, 240 per wave-tile).
//
// Round-2 change: W is staged TRANSPOSED in LDS (Wt[h][dk], stride 162) so
// each A fragment is a single ds_load_b64 into an adjacent VGPR pair,
// eliminating the v_dual_mov shuffles seen in round 1.
// ---------------------------------------------------------------------------

typedef __attribute__((ext_vector_type(2))) float v2f;
typedef __attribute__((ext_vector_type(8))) float v8f;

#define NB 16     // batch
#define NN 128    // nodes
#define NH 96     // heads total
#define KTOT 160  // 5 * 32 contraction
#define WT_STRIDE 162              // padded row stride (floats) for Wt[h][dk]
#define EIT_STRIDE 164             // padded row stride (floats) for eiT panel
#define WT_FLOATS (NH * WT_STRIDE) // 15552
#define EIT_FLOATS (16 * EIT_STRIDE)
#define SMEM_FLOATS (WT_FLOATS + 4 * EIT_FLOATS)

__global__ void __launch_bounds__(128) graph_bias_wmma(
    const float* __restrict__ attn_bias,      // [16,129,129]
    const int*   __restrict__ spatial_pos,    // [16,128,128]
    const int*   __restrict__ edge_input,     // [16,128,128,5,3]
    const float* __restrict__ edge_enc_w,     // [1537,32]
    const float* __restrict__ edge_dis_w,     // use first 160*96, [dk][h]
    const float* __restrict__ spatial_enc_w,  // [522,96]
    float*       __restrict__ out)            // [12,16,8,129,129]
{
    extern __shared__ float smem[];
    float* Wt = smem;  // [96][162]: Wt[h][dk] = W[dk][h]

    const int tid  = threadIdx.x;
    const int wave = tid >> 5;
    const int lane = tid & 31;
    float* eiT = smem + WT_FLOATS + wave * EIT_FLOATS;  // [16 pairs][164]

    // ---- stage W transposed into LDS, whole block (one-time scatter) ----
    {
        #pragma unroll
        for (int idx = tid; idx < KTOT * NH; idx += 128) {
            const int dk = idx / NH;
            const int h  = idx - dk * NH;
            Wt[h * WT_STRIDE + dk] = edge_dis_w[idx];  // coalesced global read
        }
    }

    // ---- tile decode: one pair-tile (b, n, 16 consecutive m) per wave ----
    const int tile = blockIdx.x * 4 + wave;     // 0 .. 16383
    const int b  = tile >> 10;                  // / (128*8)
    const int n  = (tile >> 3) & 127;
    const int m0 = (tile & 7) << 4;

    const int p  = lane & 15;   // pair within tile (also A-row head index)
    const int hf = lane >> 4;   // half-wave selector
    const int k0 = hf << 4;     // k range [k0, k0+16) of each 32-chunk
    const int m  = m0 + p;

    // ---- gather + sum edge encodings -> eiT[p][dk] (B-matrix = ei^T) ----
    {
        const int* ei = edge_input + (long)((b * NN + n) * NN + m) * 15;
        float* dstp = eiT + p * EIT_STRIDE + k0;
        #pragma unroll
        for (int d = 0; d < 5; ++d) {
            const int i0 = ei[d * 3 + 0];
            const int i1 = ei[d * 3 + 1];
            const int i2 = ei[d * 3 + 2];
            const float4* r0 = (const float4*)(edge_enc_w + i0 * 32 + k0);
            const float4* r1 = (const float4*)(edge_enc_w + i1 * 32 + k0);
            const float4* r2 = (const float4*)(edge_enc_w + i2 * 32 + k0);
            float4* dd = (float4*)(dstp + d * 32);
            #pragma unroll
            for (int q = 0; q < 4; ++q) {
                float4 a = r0[q], c = r1[q], e = r2[q];
                float4 s;
                s.x = a.x + c.x + e.x;
                s.y = a.y + c.y + e.y;
                s.z = a.z + c.z + e.z;
                s.w = a.w + c.w + e.w;
                dd[q] = s;   // mean /3 folded into final scale
            }
        }
    }
    __syncthreads();

    // ---- WMMA main loop: C[h_tile, m_tile] += W^T(frag) x eiT(frag) ----
    v8f acc[6] = {};                           // 6 head tiles of 16
    const float* bprow = eiT + p * EIT_STRIDE; // B frag row (pair p)
    const float* aprow = Wt + p * WT_STRIDE;   // A frag rows (head t*16 + p)
    for (int kk = 0; kk < 40; ++kk) {
        const int kb = kk * 4 + hf * 2;        // this half-wave's K pair
        v2f bf = *(const v2f*)(bprow + kb);    // one ds_load_b64
        #pragma unroll
        for (int t = 0; t < 6; ++t) {
            v2f af = *(const v2f*)(aprow + t * 16 * WT_STRIDE + kb);  // b64
            acc[t] = __builtin_amdgcn_wmma_f32_16x16x4_f32(
                false, af, false, bf, (short)0, acc[t], false, false);
        }
    }

    // ---- fused epilogue: scale by 1/(3*sp), + spatial bias, + 2*attn_bias,
    //      permuted coalesced store into (layer, b, ah, i=n+1, j=m+1) ----
    const int pidx = (b * NN + n) * NN + m;
    const int spos = spatial_pos[pidx];
    int sp = (spos == 0) ? 1 : spos;
    sp = (sp > 1) ? sp - 1 : sp;
    sp = (sp > 5) ? 5 : sp;
    const float scale = 1.0f / (3.0f * (float)sp);
    const float ab2 = 2.0f * attn_bias[b * 16641 + (n + 1) * 129 + (m + 1)];
    const float* srow = spatial_enc_w + spos * NH;
    const int outoff = (n + 1) * 129 + (m + 1);
    #pragma unroll
    for (int t = 0; t < 6; ++t) {
        #pragma unroll
        for (int r = 0; r < 8; ++r) {
            const int h = t * 16 + hf * 8 + r;   // C row M = r + 8*hf
            const float v = acc[t][r] * scale + srow[h] + ab2;
            const int layer = h >> 3, ah = h & 7;
            out[(size_t)((layer * NB + b) * 8 + ah) * 16641 + outoff] = v;
        }
    }
}

// ---- borders: i==0 row and j==0 column get 2*attn_bias + virt_w[h] ----
__global__ void graph_bias_border(const float* __restrict__ attn_bias,
                                  const float* __restrict__ virt_w,
                                  float* __restrict__ out)
{
    const int tid = blockIdx.x * blockDim.x + threadIdx.x;
    const int TOTAL = 1536 * 257;
    if (tid >= TOTAL) return;
    const int o = tid / 257;       // output plane: (layer*16 + b)*8 + ah
    const int e = tid % 257;
    const int layer = o >> 7;      // / 128
    const int b  = (o >> 3) & 15;
    const int ah = o & 7;
    const int h  = layer * 8 + ah;
    const int i = (e < 129) ? 0 : (e - 128);
    const int j = (e < 129) ? e : 0;
    const float v = 2.0f * attn_bias[b * 16641 + i * 129 + j] + virt_w[h];
    out[(size_t)o * 16641 + i * 129 + j] = v;
}

extern "C" void kernel_launch(void* const* d_in, const int* in_sizes, int n_in,
                              void* d_out, int out_size, void* d_ws, size_t ws_size,
                              hipStream_t stream) {
    const float* attn_bias     = (const float*)d_in[0];
    const int*   spatial_pos   = (const int*)d_in[1];
    // d_in[2] node_attr: only used for shapes in reference
    const int*   edge_input    = (const int*)d_in[3];
    const float* edge_enc_w    = (const float*)d_in[4];
    const float* edge_dis_w    = (const float*)d_in[5];
    const float* spatial_enc_w = (const float*)d_in[6];
    const float* virt_w        = (const float*)d_in[7];
    float* out = (float*)d_out;

    const size_t shmem = SMEM_FLOATS * sizeof(float);  // ~104 KB of 320 KB/WGP
    (void)hipFuncSetAttribute((const void*)graph_bias_wmma,
                              hipFuncAttributeMaxDynamicSharedMemorySize,
                              (int)shmem);

    // 16384 pair-tiles, 4 per block (1 per wave32)
    graph_bias_wmma<<<4096, 128, shmem, stream>>>(
        attn_bias, spatial_pos, edge_input, edge_enc_w, edge_dis_w,
        spatial_enc_w, out);

    const int border_total = 1536 * 257;
    graph_bias_border<<<(border_total + 255) / 256, 256, 0, stream>>>(
        attn_bias, virt_w, out);
}